// relu1_76467597738121
// MI455X (gfx1250) — compile-verified
//
#include <hip/hip_runtime.h>

// ReLU on 16384x8192 fp32: pure HBM-streaming problem.
// 1 GiB total traffic @ 23.3 TB/s => ~46 us floor. No matrix FLOPs => no WMMA;
// the target is the gfx1250 vector-memory path: B128 loads/stores with NT hints.
// Round-3 change: block-uniform fast path removes per-load exec masking /
// zero-init mov noise seen in the round-2 histogram (valu=80 -> expect ~30).

typedef __attribute__((ext_vector_type(4))) float v4f;  // clang vector: OK for NT builtins

static constexpr int BLOCK  = 256;           // 8 wave32s per block
static constexpr int UNROLL = 4;             // 4 float4 per thread -> 4 B128 loads in flight
static constexpr int TILE   = BLOCK * UNROLL; // float4s per block

__device__ __forceinline__ v4f relu4(v4f v) {
  v4f r;
  r.x = fmaxf(v.x, 0.0f);
  r.y = fmaxf(v.y, 0.0f);
  r.z = fmaxf(v.z, 0.0f);
  r.w = fmaxf(v.w, 0.0f);
  return r;
}

__global__ __launch_bounds__(BLOCK) void relu_nt_b128_kernel(
    const v4f* __restrict__ in, v4f* __restrict__ out, int n4) {
  const int blockBase = blockIdx.x * TILE;      // uniform
  const int t = threadIdx.x;

  if (blockBase + TILE <= n4) {
    // ---- Fast path (every block at this problem size): no guards, no exec
    // juggling. Issue all 4 NT B128 loads, then 16 v_max_num_f32, then stores.
    const v4f* pin  = in  + blockBase + t;
    v4f*       pout = out + blockBase + t;

    v4f v0 = __builtin_nontemporal_load(pin + 0 * BLOCK);
    v4f v1 = __builtin_nontemporal_load(pin + 1 * BLOCK);
    v4f v2 = __builtin_nontemporal_load(pin + 2 * BLOCK);
    v4f v3 = __builtin_nontemporal_load(pin + 3 * BLOCK);

    __builtin_nontemporal_store(relu4(v0), pout + 0 * BLOCK);
    __builtin_nontemporal_store(relu4(v1), pout + 1 * BLOCK);
    __builtin_nontemporal_store(relu4(v2), pout + 2 * BLOCK);
    __builtin_nontemporal_store(relu4(v3), pout + 3 * BLOCK);
  } else {
    // ---- Tail path (not taken for 16384x8192, kept for generality).
#pragma unroll
    for (int k = 0; k < UNROLL; ++k) {
      int idx = blockBase + k * BLOCK + t;
      if (idx < n4) {
        v4f v = __builtin_nontemporal_load(&in[idx]);
        __builtin_nontemporal_store(relu4(v), &out[idx]);
      }
    }
  }
}

extern "C" void kernel_launch(void* const* d_in, const int* in_sizes, int n_in,
                              void* d_out, int out_size, void* d_ws, size_t ws_size,
                              hipStream_t stream) {
  (void)n_in; (void)d_ws; (void)ws_size; (void)out_size;
  const v4f* in  = (const v4f*)d_in[0];
  v4f*       out = (v4f*)d_out;

  int n  = in_sizes[0];          // 134,217,728 floats
  int n4 = n >> 2;               // float4 count (exact)
  int grid = (n4 + TILE - 1) / TILE;   // 32,768 blocks

  relu_nt_b128_kernel<<<grid, BLOCK, 0, stream>>>(in, out, n4);
}